// HierarchicalKVCache_34677565948799
// MI455X (gfx1250) — compile-verified
//
#include <hip/hip_runtime.h>

// ---------------------------------------------------------------------------
// HierarchicalKVCache for MI455X (gfx1250).
// Bandwidth-bound (~0.7 GB HBM traffic => ~30us @ 23.3 TB/s).
//  - Scorer MLP GEMM (1024 x 6144 x 256) on V_WMMA_F32_16X16X4_F32 (f32 in /
//    f32 acc), activations staged in LDS, fc1_w streamed from L2 (6 MB
//    resident in the 192 MB L2).
//  - All streaming assembly kernels use 16B-per-lane (b128) accesses:
//    one wave32 per cache row, 32 lanes x 4 elements = 128 (= D).
// ---------------------------------------------------------------------------

typedef float v2f __attribute__((ext_vector_type(2)));
typedef float v8f __attribute__((ext_vector_type(8)));
typedef _Float16 h4 __attribute__((ext_vector_type(4)));

#define B_      4
#define H_      16
#define D_      128
#define NNEW    256
#define CAP1    512
#define CAP2    2048
#define CAP3    4096
#define DMODEL  2048   // H_*D_
#define HIDDEN  256
#define KDIM    6144   // 3*DMODEL
#define KEEP1   (CAP1 - NNEW)   // 256
#define KEEP2   (CAP2 - NNEW)   // 1792

// ---------------------------------------------------------------------------
// Kernel 1: scorer GEMM  y = relu(x @ fc1_w.T + b1) @ fc2_w.T + b2
// x (1024 x 6144) is the virtual concat [k_flat | v_flat | hidden].
// Grid: 16 WGs (64 M-rows each), 512 threads = 16 waves (one N-tile each),
// each wave owns 4 M-subtiles. K chunked by 128 (each chunk hits exactly one
// concat source).
// ---------------------------------------------------------------------------
__global__ __launch_bounds__(512) void scorer_gemm(
    const float* __restrict__ key, const float* __restrict__ val,
    const float* __restrict__ hid, const float* __restrict__ fc1_w,
    const float* __restrict__ fc1_b, const float* __restrict__ fc2_w,
    const float* __restrict__ fc2_b, float* __restrict__ new_scores)
{
    __shared__ float lds_x[64 * 128];     // 32 KB activation chunk
    __shared__ float lds_part[16 * 64];   // per-wave row partials

    const int tid   = threadIdx.x;
    const int wave  = tid >> 5;           // 0..15 -> N tile
    const int lane  = tid & 31;
    const int lhalf = lane >> 4;          // 0: K{0,1}, 1: K{2,3}
    const int lidx  = lane & 15;
    const int wg    = blockIdx.x;         // 0..15 -> 64-row M block

    v8f acc[4];
#pragma unroll
    for (int t = 0; t < 4; ++t)
#pragma unroll
        for (int e = 0; e < 8; ++e) acc[t][e] = 0.0f;

    const float2* wrow2 = reinterpret_cast<const float2*>(
        fc1_w + (size_t)(wave * 16 + lidx) * KDIM + lhalf * 2);

    for (int chunk = 0; chunk < 48; ++chunk) {
        __syncthreads();
        // cooperative load of 64x128 activation chunk (coalesced)
#pragma unroll
        for (int i = 0; i < 16; ++i) {
            int e   = tid + i * 512;          // 0..8191
            int row = e >> 7;                 // local M row 0..63
            int c   = e & 127;                // K within chunk
            int r   = wg * 64 + row;          // global M row
            int b   = r >> 8;                 // batch
            int nn  = r & 255;                // token
            float xv;
            if (chunk < 16) {
                xv = key[(((b * 16 + chunk) * 256 + nn) << 7) + c];
            } else if (chunk < 32) {
                xv = val[(((b * 16 + (chunk - 16)) * 256 + nn) << 7) + c];
            } else {
                xv = hid[((b << 8) + nn) * 2048 + ((chunk - 32) << 7) + c];
            }
            lds_x[e] = xv;
        }
        __syncthreads();

        const int kg0 = chunk * 128;
        for (int k = 0; k < 128; k += 4) {
            float2 bv = wrow2[(kg0 + k) >> 1];   // B[k..k+1 | k+2..k+3][n]
            v2f bm; bm[0] = bv.x; bm[1] = bv.y;
#pragma unroll
            for (int t = 0; t < 4; ++t) {
                const float2* ar = reinterpret_cast<const float2*>(
                    &lds_x[(t * 16 + lidx) * 128 + k + lhalf * 2]);
                float2 av = ar[0];
                v2f am; am[0] = av.x; am[1] = av.y;
                acc[t] = __builtin_amdgcn_wmma_f32_16x16x4_f32(
                    false, am, false, bm, (short)0, acc[t], false, false);
            }
        }
    }

    // Fused bias + relu + fc2 reduction (deterministic: shfl tree within the
    // 16-lane N group, then fixed-order sum across the 16 waves).
    __syncthreads();
    const int n   = wave * 16 + lidx;
    const float bias = fc1_b[n];
    const float w2   = fc2_w[n];
#pragma unroll
    for (int t = 0; t < 4; ++t) {
#pragma unroll
        for (int v = 0; v < 8; ++v) {
            float y = acc[t][v] + bias;
            y = fmaxf(y, 0.0f) * w2;
            y += __shfl_xor(y, 1, 32);
            y += __shfl_xor(y, 2, 32);
            y += __shfl_xor(y, 4, 32);
            y += __shfl_xor(y, 8, 32);
            if (lidx == 0)
                lds_part[wave * 64 + t * 16 + lhalf * 8 + v] = y;
        }
    }
    __syncthreads();
    if (tid < 64) {
        float sum = fc2_b[0];
        for (int w = 0; w < 16; ++w) sum += lds_part[w * 64 + tid];
        new_scores[wg * 64 + tid] = sum;   // row = b*256 + nn
    }
}

// ---------------------------------------------------------------------------
// Stable descending argsort via rank counting (matches jnp.argsort(-s)).
// ---------------------------------------------------------------------------
__global__ void rank_t1(const float* __restrict__ scores, int* __restrict__ order)
{
    __shared__ float s[CAP1];
    const int b = blockIdx.x, i = threadIdx.x;
    s[i] = scores[b * CAP1 + i];
    __syncthreads();
    const float si = s[i];
    int rank = 0;
    for (int j = 0; j < CAP1; ++j) {
        float sj = s[j];
        rank += (sj > si) || (sj == si && j < i);
    }
    order[b * CAP1 + rank] = i;
}

__global__ void rank_t2(const _Float16* __restrict__ scores, int* __restrict__ order)
{
    __shared__ float s[CAP2];
    const int b = blockIdx.x;
    for (int i = threadIdx.x; i < CAP2; i += 1024)
        s[i] = (float)scores[b * CAP2 + i];
    __syncthreads();
#pragma unroll
    for (int ii = 0; ii < 2; ++ii) {
        const int i = threadIdx.x + ii * 1024;
        const float si = s[i];
        int rank = 0;
        for (int j = 0; j < CAP2; ++j) {
            float sj = s[j];
            rank += (sj > si) || (sj == si && j < i);
        }
        order[b * CAP2 + rank] = i;
    }
}

// ---------------------------------------------------------------------------
// Tier-1 assembly: keep top KEEP1 (in sort order) then append new K/V.
// 256-thread blocks = 8 waves; one wave per (b,h,i) row; 16B per lane.
// ---------------------------------------------------------------------------
__global__ __launch_bounds__(256) void assemble_t1(
    const float* __restrict__ t1_k, const float* __restrict__ t1_v,
    const float* __restrict__ t1_s, const float* __restrict__ key,
    const float* __restrict__ val, const float* __restrict__ new_scores,
    const int* __restrict__ order1,
    float* __restrict__ ok, float* __restrict__ ov, float* __restrict__ os)
{
    const int row  = blockIdx.x * 8 + (threadIdx.x >> 5);
    const int lane = threadIdx.x & 31;
    const int i = row & (CAP1 - 1);
    const int h = (row >> 9) & 15;
    const int b = row >> 13;
    const long bh = b * 16 + h;
    const long dst = ((bh * CAP1 + i) * D_ + lane * 4) >> 2;  // float4 index

    float4 kv, vv;
    if (i < KEEP1) {
        const int src = order1[b * CAP1 + i];
        const long sidx = ((bh * CAP1 + src) * D_ + lane * 4) >> 2;
        kv = reinterpret_cast<const float4*>(t1_k)[sidx];
        vv = reinterpret_cast<const float4*>(t1_v)[sidx];
        if (h == 0 && lane == 0) os[b * CAP1 + i] = t1_s[b * CAP1 + src];
    } else {
        const int j = i - KEEP1;
        const long sidx = ((bh * NNEW + j) * D_ + lane * 4) >> 2;
        kv = reinterpret_cast<const float4*>(key)[sidx];
        vv = reinterpret_cast<const float4*>(val)[sidx];
        if (h == 0 && lane == 0) os[b * CAP1 + i] = new_scores[b * NNEW + j];
    }
    reinterpret_cast<float4*>(ok)[dst] = kv;
    reinterpret_cast<float4*>(ov)[dst] = vv;
}

// ---------------------------------------------------------------------------
// Tier-2 assembly: keep top KEEP2 of tier-2, append tier-1 evictees (as f16).
// f16-valued outputs are emitted as f32 rounded through _Float16.
// One wave per row: 4x f16 (8B) or float4 loads, float4 stores.
// ---------------------------------------------------------------------------
__global__ __launch_bounds__(256) void assemble_t2(
    const _Float16* __restrict__ t2_k, const _Float16* __restrict__ t2_v,
    const _Float16* __restrict__ t2_s, const float* __restrict__ t1_k,
    const float* __restrict__ t1_v, const float* __restrict__ t1_s,
    const int* __restrict__ order1, const int* __restrict__ order2,
    float* __restrict__ ok, float* __restrict__ ov, float* __restrict__ os)
{
    const int row  = blockIdx.x * 8 + (threadIdx.x >> 5);
    const int lane = threadIdx.x & 31;
    const int i = row & (CAP2 - 1);
    const int h = (row >> 11) & 15;
    const int b = row >> 15;
    const long bh = b * 16 + h;
    const long dst = ((bh * CAP2 + i) * D_ + lane * 4) >> 2;

    float4 kv, vv;
    if (i < KEEP2) {
        const int src = order2[b * CAP2 + i];
        const long sidx = ((bh * CAP2 + src) * D_ + lane * 4) >> 2;  // h4 index
        h4 hk = reinterpret_cast<const h4*>(t2_k)[sidx];
        h4 hv = reinterpret_cast<const h4*>(t2_v)[sidx];
        kv = make_float4((float)hk[0], (float)hk[1], (float)hk[2], (float)hk[3]);
        vv = make_float4((float)hv[0], (float)hv[1], (float)hv[2], (float)hv[3]);
        if (h == 0 && lane == 0) os[b * CAP2 + i] = (float)t2_s[b * CAP2 + src];
    } else {
        const int j = i - KEEP2;
        const int src = order1[b * CAP1 + KEEP1 + j];   // tier-1 evictee
        const long sidx = ((bh * CAP1 + src) * D_ + lane * 4) >> 2;
        float4 fk = reinterpret_cast<const float4*>(t1_k)[sidx];
        float4 fv = reinterpret_cast<const float4*>(t1_v)[sidx];
        kv = make_float4((float)(_Float16)fk.x, (float)(_Float16)fk.y,
                         (float)(_Float16)fk.z, (float)(_Float16)fk.w);
        vv = make_float4((float)(_Float16)fv.x, (float)(_Float16)fv.y,
                         (float)(_Float16)fv.z, (float)(_Float16)fv.w);
        if (h == 0 && lane == 0)
            os[b * CAP2 + i] = (float)(_Float16)t1_s[b * CAP1 + src];
    }
    reinterpret_cast<float4*>(ok)[dst] = kv;
    reinterpret_cast<float4*>(ov)[dst] = vv;
}

// ---------------------------------------------------------------------------
// Tier-3 assembly: dequantize int8(stored int32)*scale everywhere, except the
// [t3_n, t3_n+NNEW) window which receives requantized tier-2 evictees.
// One wave per row; int4/float4 per lane; absmax via in-wave shfl_xor tree.
// rintf == round-half-even == jnp.round.
// ---------------------------------------------------------------------------
__global__ __launch_bounds__(256) void assemble_t3(
    const int* __restrict__ t3_k, const int* __restrict__ t3_v,
    const float* __restrict__ sk, const float* __restrict__ sv,
    const _Float16* __restrict__ t3_s,
    const _Float16* __restrict__ t2_k, const _Float16* __restrict__ t2_v,
    const _Float16* __restrict__ t2_s, const int* __restrict__ order2,
    const int* __restrict__ t3_n_p,
    float* __restrict__ ok, float* __restrict__ ov, float* __restrict__ os)
{
    const int row  = blockIdx.x * 8 + (threadIdx.x >> 5);
    const int lane = threadIdx.x & 31;
    const int s = row & (CAP3 - 1);
    const int h = (row >> 12) & 15;
    const int b = row >> 16;
    const long bh = b * 16 + h;
    const int t3n = t3_n_p[0];
    const long dst = ((bh * CAP3 + s) * D_ + lane * 4) >> 2;

    if (s >= t3n && s < t3n + NNEW) {                 // uniform per wave
        const int j = s - t3n;
        const int src = order2[b * CAP2 + KEEP2 + j]; // tier-2 evictee
        const long sidx = ((bh * CAP2 + src) * D_ + lane * 4) >> 2;
        h4 hk = reinterpret_cast<const h4*>(t2_k)[sidx];
        h4 hv = reinterpret_cast<const h4*>(t2_v)[sidx];
        float xk[4], xv[4];
        float ak = 0.0f, av = 0.0f;
#pragma unroll
        for (int c = 0; c < 4; ++c) {
            xk[c] = (float)hk[c];
            xv[c] = (float)hv[c];
            ak = fmaxf(ak, fabsf(xk[c]));
            av = fmaxf(av, fabsf(xv[c]));
        }
#pragma unroll
        for (int off = 16; off > 0; off >>= 1) {      // wave32 absmax
            ak = fmaxf(ak, __shfl_xor(ak, off, 32));
            av = fmaxf(av, __shfl_xor(av, off, 32));
        }
        const float sck = fmaxf(ak, 1e-5f) / 127.0f;
        const float scv = fmaxf(av, 1e-5f) / 127.0f;
        float4 okv, ovv;
        float* pk = &okv.x;
        float* pv = &ovv.x;
#pragma unroll
        for (int c = 0; c < 4; ++c) {
            pk[c] = fminf(fmaxf(rintf(xk[c] / sck), -128.0f), 127.0f) * sck;
            pv[c] = fminf(fmaxf(rintf(xv[c] / scv), -128.0f), 127.0f) * scv;
        }
        reinterpret_cast<float4*>(ok)[dst] = okv;
        reinterpret_cast<float4*>(ov)[dst] = ovv;
        if (h == 0 && lane == 0)
            os[b * CAP3 + s] = (float)t2_s[b * CAP2 + src];
    } else {
        int4 qk = reinterpret_cast<const int4*>(t3_k)[dst];
        int4 qv = reinterpret_cast<const int4*>(t3_v)[dst];
        const float sck = sk[bh * CAP3 + s];
        const float scv = sv[bh * CAP3 + s];
        float4 okv = make_float4((float)qk.x * sck, (float)qk.y * sck,
                                 (float)qk.z * sck, (float)qk.w * sck);
        float4 ovv = make_float4((float)qv.x * scv, (float)qv.y * scv,
                                 (float)qv.z * scv, (float)qv.w * scv);
        reinterpret_cast<float4*>(ok)[dst] = okv;
        reinterpret_cast<float4*>(ov)[dst] = ovv;
        if (h == 0 && lane == 0)
            os[b * CAP3 + s] = (float)t3_s[b * CAP3 + s];
    }
}

// ---------------------------------------------------------------------------
extern "C" void kernel_launch(void* const* d_in, const int* in_sizes, int n_in,
                              void* d_out, int out_size, void* d_ws, size_t ws_size,
                              hipStream_t stream)
{
    const float*    key    = (const float*)   d_in[0];
    const float*    value  = (const float*)   d_in[1];
    const float*    hidden = (const float*)   d_in[2];
    const float*    fc1_w  = (const float*)   d_in[3];
    const float*    fc1_b  = (const float*)   d_in[4];
    const float*    fc2_w  = (const float*)   d_in[5];
    const float*    fc2_b  = (const float*)   d_in[6];
    const float*    t1_k   = (const float*)   d_in[7];
    const float*    t1_v   = (const float*)   d_in[8];
    const float*    t1_s   = (const float*)   d_in[9];
    const _Float16* t2_k   = (const _Float16*)d_in[10];
    const _Float16* t2_v   = (const _Float16*)d_in[11];
    const _Float16* t2_s   = (const _Float16*)d_in[12];
    const int*      t3_k   = (const int*)     d_in[13];
    const int*      t3_v   = (const int*)     d_in[14];
    const float*    t3_sk  = (const float*)   d_in[15];
    const float*    t3_sv  = (const float*)   d_in[16];
    const _Float16* t3_s   = (const _Float16*)d_in[17];
    const int*      t3_n   = (const int*)     d_in[18];

    // Output layout: tuple concatenated flat, everything as f32.
    float* o     = (float*)d_out;
    float* o_t1k = o;                                   // B*H*CAP1*D
    float* o_t1v = o_t1k + (size_t)B_ * H_ * CAP1 * D_;
    float* o_t1s = o_t1v + (size_t)B_ * H_ * CAP1 * D_; // B*CAP1
    float* o_t2k = o_t1s + (size_t)B_ * CAP1;
    float* o_t2v = o_t2k + (size_t)B_ * H_ * CAP2 * D_;
    float* o_t2s = o_t2v + (size_t)B_ * H_ * CAP2 * D_; // B*CAP2
    float* o_t3k = o_t2s + (size_t)B_ * CAP2;
    float* o_t3v = o_t3k + (size_t)B_ * H_ * CAP3 * D_;
    float* o_t3s = o_t3v + (size_t)B_ * H_ * CAP3 * D_; // B*CAP3

    // Workspace: new_scores[1024] | order1[4*512] | order2[4*2048]
    float* ws_scores = (float*)d_ws;
    int*   ws_order1 = (int*)(ws_scores + B_ * NNEW);
    int*   ws_order2 = ws_order1 + B_ * CAP1;

    scorer_gemm<<<16, 512, 0, stream>>>(key, value, hidden, fc1_w, fc1_b,
                                        fc2_w, fc2_b, ws_scores);
    rank_t1<<<B_, CAP1, 0, stream>>>(t1_s, ws_order1);
    rank_t2<<<B_, 1024, 0, stream>>>(t2_s, ws_order2);

    assemble_t1<<<B_ * H_ * CAP1 / 8, 256, 0, stream>>>(
        t1_k, t1_v, t1_s, key, value, ws_scores, ws_order1,
        o_t1k, o_t1v, o_t1s);
    assemble_t2<<<B_ * H_ * CAP2 / 8, 256, 0, stream>>>(
        t2_k, t2_v, t2_s, t1_k, t1_v, t1_s, ws_order1, ws_order2,
        o_t2k, o_t2v, o_t2s);
    assemble_t3<<<B_ * H_ * CAP3 / 8, 256, 0, stream>>>(
        t3_k, t3_v, t3_sk, t3_sv, t3_s, t2_k, t2_v, t2_s, ws_order2, t3_n,
        o_t3k, o_t3v, o_t3s);
}